// BidirectionalEncoder_34626026341076
// MI455X (gfx1250) — compile-verified
//
#include <hip/hip_runtime.h>
#include <hip/hip_bf16.h>

#define S_LEN 4096
#define I_DIM 1024
#define H_DIM 1024
#define G4    (4 * H_DIM)          // 4096 gate rows
#define WROW  (H_DIM + I_DIM)      // 2048 columns per W row

typedef float v2f __attribute__((ext_vector_type(2)));
typedef float v8f __attribute__((ext_vector_type(8)));

// ---------------------------------------------------------------------------
// GEMM tiling parameters
// ---------------------------------------------------------------------------
#define KC    32                   // K-chunk staged per double-buffer step
#define APAD  36                   // padded row stride (floats): 144B, 16B-aligned,
                                   // conflict-free ds_load_b64 fragment reads
#define SM_BLK 128                 // block tile: 128 sequence rows
#define GN_BLK 256                 // block tile: 256 gate cols
#define A_FLOATS (SM_BLK * APAD)   // 4608
#define B_FLOATS (GN_BLK * APAD)   // 9216
// dynamic LDS layout: [A0][A1][B0][B1]
#define LDS_BYTES ((2 * A_FLOATS + 2 * B_FLOATS) * 4)   // 110592

// 16B async copy from global address g to LDS address lds_off, tracked by ASYNCcnt
__device__ __forceinline__ void async_copy16(unsigned lds_off, const float* g) {
    unsigned long long ga = (unsigned long long)g;
    asm volatile("global_load_async_to_lds_b128 %0, %1, off"
                 :: "v"(lds_off), "v"(ga) : "memory");
}
__device__ __forceinline__ void wait_async_all() {
    asm volatile("s_wait_asynccnt 0x0" ::: "memory");
}

// ---------------------------------------------------------------------------
// Phase 1: x_proj[d] = inputs @ Wx[d]^T + b[d]   (fp32 WMMA 16x16x4)
// Tiles staged through LDS via async-to-LDS DMA, double-buffered so the DMA of
// chunk c+1 overlaps the WMMA chain of chunk c.
// ---------------------------------------------------------------------------
__global__ __launch_bounds__(256)
void xproj_gemm_kernel(const float* __restrict__ inputs,
                       const float* __restrict__ Wf, const float* __restrict__ bf,
                       const float* __restrict__ Wb, const float* __restrict__ bwd_bias,
                       float* __restrict__ xpf, float* __restrict__ xpb)
{
    extern __shared__ float smem[];   // [A0][A1][B0][B1]

    const int dir = blockIdx.z;
    const float* __restrict__ W    = dir ? Wb       : Wf;
    const float* __restrict__ bias = dir ? bwd_bias : bf;
    float* __restrict__ xp         = dir ? xpb      : xpf;

    const int tid    = threadIdx.x;
    const int lane   = tid & 31;
    const int wave   = tid >> 5;
    const int gn_blk = blockIdx.x * GN_BLK;
    const int sm_blk = blockIdx.y * SM_BLK;
    const int gnw    = (wave & 3) * 64;     // wave tile inside block (g)
    const int smw    = (wave >> 2) * 64;    // wave tile inside block (s)

    const int lrow  = lane & 15;
    const int khalf = lane >> 4;

    v8f acc[4][4];
    #pragma unroll
    for (int mi = 0; mi < 4; ++mi)
        #pragma unroll
        for (int ni = 0; ni < 4; ++ni)
            #pragma unroll
            for (int r = 0; r < 8; ++r) acc[mi][ni][r] = 0.0f;

    // --- stage one KC-chunk of A (128x32) and B (256x32) into LDS buffer `buf`
    auto stage = [&](int kc, int buf) {
        float* As = smem + buf * A_FLOATS;                // runtime address math only
        float* Bs = smem + 2 * A_FLOATS + buf * B_FLOATS;
        // A: 1024 16B pieces (128 rows x 8), 4 per thread
        #pragma unroll
        for (int q = 0; q < 4; ++q) {
            const int p   = tid + 256 * q;
            const int row = p >> 3, col = p & 7;
            const float* g = inputs + (size_t)(sm_blk + row) * I_DIM + kc + col * 4;
            unsigned l = (unsigned)(size_t)(As + row * APAD + col * 4);
            async_copy16(l, g);
        }
        // B: 2048 16B pieces (256 rows x 8), 8 per thread
        #pragma unroll
        for (int q = 0; q < 8; ++q) {
            const int p   = tid + 256 * q;
            const int row = p >> 3, col = p & 7;
            const float* g = W + (size_t)(gn_blk + row) * WROW + H_DIM + kc + col * 4;
            unsigned l = (unsigned)(size_t)(Bs + row * APAD + col * 4);
            async_copy16(l, g);
        }
    };

    stage(0, 0);

    const int NCHUNK = I_DIM / KC;   // 32
    for (int c = 0; c < NCHUNK; ++c) {
        wait_async_all();            // our chunk-c writes landed
        __syncthreads();             // everyone's landed; prior reads finished
        if (c + 1 < NCHUNK) stage((c + 1) * KC, (c + 1) & 1);

        const float* Ab = smem + (c & 1) * A_FLOATS;
        const float* Bb = smem + 2 * A_FLOATS + (c & 1) * B_FLOATS;
        #pragma unroll
        for (int ks = 0; ks < KC; ks += 4) {
            const int kb = ks + khalf * 2;
            v2f afrag[4], bfrag[4];
            #pragma unroll
            for (int mi = 0; mi < 4; ++mi)
                afrag[mi] = *(const v2f*)(Ab + (smw + mi * 16 + lrow) * APAD + kb);
            #pragma unroll
            for (int ni = 0; ni < 4; ++ni)
                bfrag[ni] = *(const v2f*)(Bb + (gnw + ni * 16 + lrow) * APAD + kb);
            #pragma unroll
            for (int mi = 0; mi < 4; ++mi)
                #pragma unroll
                for (int ni = 0; ni < 4; ++ni)
                    acc[mi][ni] = __builtin_amdgcn_wmma_f32_16x16x4_f32(
                        false, afrag[mi], false, bfrag[ni],
                        (short)0, acc[mi][ni], false, false);
        }
        __syncthreads();             // reads of buf (c&1) done before chunk c+2 rewrites it
    }

    // Epilogue: +bias, store. C/D layout: VGPR r -> M = r + (lane>>4)*8, N = lane&15.
    const int ncol  = lane & 15;
    const int rbase = (lane >> 4) * 8;
    #pragma unroll
    for (int ni = 0; ni < 4; ++ni) {
        const int g  = gn_blk + gnw + ni * 16 + ncol;
        const float bv = bias[g];
        #pragma unroll
        for (int mi = 0; mi < 4; ++mi)
            #pragma unroll
            for (int r = 0; r < 8; ++r) {
                const int m = sm_blk + smw + mi * 16 + rbase + r;
                xp[(size_t)m * G4 + g] = acc[mi][ni][r] + bv;
            }
    }
}

// ---------------------------------------------------------------------------
// Phase 2: persistent bidirectional LSTM recurrence.
// 64 blocks (32 per direction), each owns 32 hidden units (128 gate rows).
// Wh stays L2-resident (16.8MB/dir << 192MB L2). Grid barrier per step.
// ---------------------------------------------------------------------------
#define NBLK_PER_DIR 32
#define HS_PER_BLK   (H_DIM / NBLK_PER_DIR)   // 32

__device__ __forceinline__ float sigmoidf_(float x) {
    return 1.0f / (1.0f + __expf(-x));
}

__device__ __forceinline__ float wave32_reduce_add(float v) {
    v += __int_as_float(__builtin_amdgcn_ds_swizzle(__float_as_int(v), 0x401f)); // xor 16
    v += __int_as_float(__builtin_amdgcn_ds_swizzle(__float_as_int(v), 0x201f)); // xor 8
    v += __int_as_float(__builtin_amdgcn_ds_swizzle(__float_as_int(v), 0x101f)); // xor 4
    v += __int_as_float(__builtin_amdgcn_ds_swizzle(__float_as_int(v), 0x081f)); // xor 2
    v += __int_as_float(__builtin_amdgcn_ds_swizzle(__float_as_int(v), 0x041f)); // xor 1
    return v;
}

__global__ __launch_bounds__(256)
void lstm_persistent_kernel(const float* __restrict__ Wf, const float* __restrict__ Wb,
                            const float* __restrict__ xpf, const float* __restrict__ xpb,
                            float* __restrict__ hbuf,      // [2 dir][2 parity][H]
                            unsigned* __restrict__ syncv,  // [0]=counter [1]=generation
                            float* __restrict__ out)       // (S,2H) all_hidden ++ (2H) final
{
    const int tid     = threadIdx.x;
    const int dir     = blockIdx.x >> 5;      // 0 = fwd, 1 = bwd
    const int blk     = blockIdx.x & 31;
    const unsigned nb = gridDim.x;

    const float* __restrict__ W  = dir ? Wb  : Wf;
    const float* __restrict__ xp = dir ? xpb : xpf;
    float* __restrict__ hb       = hbuf + (size_t)dir * 2 * H_DIM;

    const int u0   = blk * HS_PER_BLK;
    const int lane = tid & 31;
    const int wave = tid >> 5;

    __shared__ float h_s[H_DIM];
    __shared__ float z_s[4 * HS_PER_BLK];
    __shared__ float c_s[HS_PER_BLK];
    if (tid < HS_PER_BLK) c_s[tid] = 0.0f;

    for (int t = 0; t < S_LEN; ++t) {
        const int xidx = dir ? (S_LEN - 1 - t) : t;

        const float* hsrc = hb + (size_t)(t & 1) * H_DIM;
        for (int j = tid; j < H_DIM; j += 256) h_s[j] = hsrc[j];
        __syncthreads();

        for (int i = 0; i < 16; ++i) {
            const int lr   = wave * 16 + i;          // 0..127
            const int gate = lr >> 5;
            const int unit = lr & 31;
            const int grow = gate * H_DIM + u0 + unit;
            const float* __restrict__ wrow = W + (size_t)grow * WROW; // Wh = first H cols
            float acc = 0.0f;
            #pragma unroll
            for (int kk = 0; kk < H_DIM / 128; ++kk) {
                const int j = kk * 128 + lane * 4;
                const float4 wv = *(const float4*)(wrow + j);
                acc += wv.x * h_s[j]     + wv.y * h_s[j + 1]
                     + wv.z * h_s[j + 2] + wv.w * h_s[j + 3];
            }
            acc = wave32_reduce_add(acc);
            if (lane == 0)
                z_s[lr] = acc + xp[(size_t)xidx * G4 + grow];
        }
        __syncthreads();

        if (tid < HS_PER_BLK) {
            const int u = tid;
            const float f  = sigmoidf_(z_s[u]);
            const float ig = sigmoidf_(z_s[32 + u]);
            const float g  = tanhf(z_s[64 + u]);
            const float o  = sigmoidf_(z_s[96 + u]);
            const float c  = f * c_s[u] + ig * g;
            c_s[u] = c;
            const float h = o * tanhf(c);
            hb[(size_t)((t + 1) & 1) * H_DIM + u0 + u] = h;
            out[(size_t)xidx * (2 * H_DIM) + (size_t)dir * H_DIM + u0 + u] = h;
            if (t == S_LEN - 1)
                out[(size_t)S_LEN * (2 * H_DIM) + (size_t)dir * H_DIM + u0 + u] = h;
        }

        __syncthreads();
        if (tid == 0) {
            __threadfence();
            const unsigned arr = __hip_atomic_fetch_add(&syncv[0], 1u,
                                    __ATOMIC_ACQ_REL, __HIP_MEMORY_SCOPE_AGENT);
            if (arr == nb - 1u) {
                __hip_atomic_store(&syncv[0], 0u, __ATOMIC_RELAXED, __HIP_MEMORY_SCOPE_AGENT);
                __hip_atomic_fetch_add(&syncv[1], 1u, __ATOMIC_RELEASE, __HIP_MEMORY_SCOPE_AGENT);
            } else {
                while (__hip_atomic_load(&syncv[1], __ATOMIC_ACQUIRE,
                                         __HIP_MEMORY_SCOPE_AGENT) < (unsigned)(t + 1))
                    __builtin_amdgcn_s_sleep(2);
            }
        }
        __syncthreads();
    }
}

// ---------------------------------------------------------------------------
// Host-side launch. Workspace layout (bytes):
//   [0,256)                : sync counters
//   [256, 256+16K)         : h double buffers (2 dir x 2 parity x 1024 f32)
//   [16640, 16640+64MB)    : x_proj forward  (S x 4H f32)
//   [+64MB, +128MB)        : x_proj backward
// ---------------------------------------------------------------------------
extern "C" void kernel_launch(void* const* d_in, const int* in_sizes, int n_in,
                              void* d_out, int out_size, void* d_ws, size_t ws_size,
                              hipStream_t stream) {
    const float* inputs = (const float*)d_in[0];
    const float* fwd_W  = (const float*)d_in[1];
    const float* fwd_b  = (const float*)d_in[2];
    const float* bwd_W  = (const float*)d_in[3];
    const float* bwd_b  = (const float*)d_in[4];
    float* out = (float*)d_out;

    unsigned char* ws = (unsigned char*)d_ws;
    unsigned* syncv = (unsigned*)ws;
    float* hbuf = (float*)(ws + 256);
    float* xpf  = (float*)(ws + 16640);
    float* xpb  = xpf + (size_t)S_LEN * G4;

    (void)hipMemsetAsync(ws, 0, 16640, stream);

    dim3 ggrid(G4 / GN_BLK, S_LEN / SM_BLK, 2);   // 16 x 32 x 2
    xproj_gemm_kernel<<<ggrid, 256, LDS_BYTES, stream>>>(inputs, fwd_W, fwd_b,
                                                         bwd_W, bwd_b, xpf, xpb);

    lstm_persistent_kernel<<<2 * NBLK_PER_DIR, 256, 0, stream>>>(fwd_W, bwd_W, xpf, xpb,
                                                                 hbuf, syncv, out);
}